// GATRockPredictor_26104811225565
// MI455X (gfx1250) — compile-verified
//
#include <hip/hip_runtime.h>
#include <hip/hip_bf16.h>

#define NNODES 100000
#define NEDGES 1600000
#define INCH   32
#define HIDC   64
#define NHEADS 2
#define HCAT   128           // HEADS*HID
#define NCLS   12
#define NEGSLOPE 0.2f

typedef __attribute__((ext_vector_type(2))) float v2f;
typedef __attribute__((ext_vector_type(8))) float v8f;

__device__ __forceinline__ float leakyr(float x) { return x > 0.f ? x : NEGSLOPE * x; }

// float atomic max via int-ordering trick (init bytes 0xAA/0xFF both behave as -inf-ish;
// we memset to 0xFF: as signed int it's -1 (< any positive-float bits), as unsigned it's
// UINT_MAX (> any negative-float bits), so both branches converge correctly).
__device__ __forceinline__ void atomicMaxF(float* addr, float v) {
    if (v >= 0.f) atomicMax((int*)addr, __float_as_int(v));
    else          atomicMin((unsigned int*)addr, __float_as_uint(v));
}

// ---------------------------------------------------------------------------
// FP32 WMMA GEMM: D[nrows,M] = A[nrows,K] @ B[K,M] (+ bias[M] if non-null).
// One wave -> one 16-row tile x full M width (MT 16-col tiles).
// A frag (16x4 f32): lane&15 = row, lanes>=16 hold K=2,3.  B frag symmetric.
// C/D: VGPR r holds row (r + 8*(lane>>4)), col = lane&15.
// ---------------------------------------------------------------------------
template <int K, int M, int MT>
__global__ __launch_bounds__(256) void gemm_wmma_f32(
    const float* __restrict__ A, const float* __restrict__ B,
    const float* __restrict__ bias, float* __restrict__ D, int nrows)
{
    const int wave = threadIdx.x >> 5;
    const int lane = threadIdx.x & 31;
    const int tile = blockIdx.x * 8 + wave;
    const int ntiles = nrows >> 4;            // nrows is a multiple of 16
    if (tile >= ntiles) return;               // wave-uniform: EXEC stays all-1 for WMMA
    const int half = lane >> 4;
    const int mrow = lane & 15;
    const int ncol = lane & 15;
    const int m0 = tile << 4;

    v8f acc[MT] = {};

    const float* __restrict__ arow = A + (size_t)(m0 + mrow) * K;
    for (int kb = 0; kb < K; kb += 4) {
        const int k0 = kb + 2 * half;
        v2f a;
        a.x = arow[k0];
        a.y = arow[k0 + 1];
#pragma unroll
        for (int t = 0; t < MT; ++t) {
            const int n = t * 16 + ncol;
            v2f b;
            if ((M & 15) == 0 || n < M) {
                b.x = B[(size_t)k0 * M + n];
                b.y = B[(size_t)(k0 + 1) * M + n];
            } else {
                b.x = 0.f; b.y = 0.f;
            }
            acc[t] = __builtin_amdgcn_wmma_f32_16x16x4_f32(
                false, a, false, b, (short)0, acc[t], false, false);
        }
    }

#pragma unroll
    for (int t = 0; t < MT; ++t) {
        const int n = t * 16 + ncol;
        if ((M & 15) != 0 && n >= M) continue;
        const float bv = bias ? bias[n] : 0.f;
#pragma unroll
        for (int r = 0; r < 8; ++r) {
            const int m = m0 + r + 8 * half;
            D[(size_t)m * M + n] = acc[t][r] + bv;
        }
    }
}

// ---------------------------------------------------------------------------
// Per-(node,head) attention logits: als/ald[n*2+h] = sum_c H[n,h,c]*a_{s,d}[h,c]
// ---------------------------------------------------------------------------
__global__ __launch_bounds__(256) void attn_logits_kernel(
    const float* __restrict__ H, const float* __restrict__ a_s,
    const float* __restrict__ a_d, float* __restrict__ als, float* __restrict__ ald)
{
    const int idx = blockIdx.x * blockDim.x + threadIdx.x;
    if (idx >= NNODES * NHEADS) return;
    const int head = idx & 1;
    const int node = idx >> 1;
    const float4* hv = (const float4*)(H + (size_t)node * HCAT + head * HIDC);
    const float4* sv = (const float4*)(a_s + head * HIDC);
    const float4* dv = (const float4*)(a_d + head * HIDC);
    float s = 0.f, d = 0.f;
#pragma unroll
    for (int i = 0; i < HIDC / 4; ++i) {
        const float4 h4 = hv[i], s4 = sv[i], d4 = dv[i];
        s += h4.x * s4.x + h4.y * s4.y + h4.z * s4.z + h4.w * s4.w;
        d += h4.x * d4.x + h4.y * d4.y + h4.z * d4.z + h4.w * d4.w;
    }
    als[idx] = s;
    ald[idx] = d;
}

__device__ __forceinline__ void edge_sd(const int* __restrict__ ei, int e, int& s, int& d) {
    if (e < NEDGES) { s = ei[e]; d = ei[NEDGES + e]; }
    else            { s = d = e - NEDGES; }          // implicit self-loops
}

// Phase A: segment max of leaky-relu logits over destinations.
__global__ __launch_bounds__(256) void edge_max_kernel(
    const int* __restrict__ ei, const float* __restrict__ als,
    const float* __restrict__ ald, float* __restrict__ maxb)
{
    const int e = blockIdx.x * blockDim.x + threadIdx.x;
    if (e >= NEDGES + NNODES) return;
    int s, d; edge_sd(ei, e, s, d);
    atomicMaxF(&maxb[d * 2 + 0], leakyr(als[s * 2 + 0] + ald[d * 2 + 0]));
    atomicMaxF(&maxb[d * 2 + 1], leakyr(als[s * 2 + 1] + ald[d * 2 + 1]));
}

// Phase B: ex = exp(e - max[dst]); stash per-edge, accumulate denominator.
__global__ __launch_bounds__(256) void edge_exp_kernel(
    const int* __restrict__ ei, const float* __restrict__ als,
    const float* __restrict__ ald, const float* __restrict__ maxb,
    float* __restrict__ exbuf, float* __restrict__ denom)
{
    const int e = blockIdx.x * blockDim.x + threadIdx.x;
    if (e >= NEDGES + NNODES) return;
    int s, d; edge_sd(ei, e, s, d);
    const float x0 = __expf(leakyr(als[s * 2 + 0] + ald[d * 2 + 0]) - maxb[d * 2 + 0]);
    const float x1 = __expf(leakyr(als[s * 2 + 1] + ald[d * 2 + 1]) - maxb[d * 2 + 1]);
    exbuf[(size_t)e * 2 + 0] = x0;
    exbuf[(size_t)e * 2 + 1] = x1;
    atomicAdd(&denom[d * 2 + 0], x0);
    atomicAdd(&denom[d * 2 + 1], x1);
}

// Phase C: one wave per edge; lane covers 4 channels; scatter alpha*h[src] into AGG[dst].
__global__ __launch_bounds__(256) void edge_aggr_kernel(
    const int* __restrict__ ei, const float* __restrict__ H,
    const float* __restrict__ exbuf, const float* __restrict__ denom,
    float* __restrict__ AGG)
{
    const int e = blockIdx.x * 8 + (threadIdx.x >> 5);
    if (e >= NEDGES + NNODES) return;
    const int lane = threadIdx.x & 31;
    int s, d; edge_sd(ei, e, s, d);
    const int c = lane * 4;                // 0..124, head boundary at 64
    const int head = c >> 6;
    const float alpha = exbuf[(size_t)e * 2 + head] / denom[d * 2 + head];
    const float4 h4 = *(const float4*)(H + (size_t)s * HCAT + c);
    float* out = AGG + (size_t)d * HCAT + c;
    atomicAdd(out + 0, h4.x * alpha);
    atomicAdd(out + 1, h4.y * alpha);
    atomicAdd(out + 2, h4.z * alpha);
    atomicAdd(out + 3, h4.w * alpha);
}

// concat=True finalize: OUT = prelu(AGG + bias)
__global__ __launch_bounds__(256) void finalize_concat_kernel(
    const float* __restrict__ AGG, const float* __restrict__ bias,
    const float* __restrict__ prelu_a, float* __restrict__ OUT)
{
    const int idx = blockIdx.x * blockDim.x + threadIdx.x;   // float4 units
    if (idx >= NNODES * HCAT / 4) return;
    const int k = (idx & (HCAT / 4 - 1)) * 4;
    const float a = *prelu_a;
    float4 v = ((const float4*)AGG)[idx];
    v.x += bias[k + 0]; v.y += bias[k + 1]; v.z += bias[k + 2]; v.w += bias[k + 3];
    v.x = v.x >= 0.f ? v.x : a * v.x;
    v.y = v.y >= 0.f ? v.y : a * v.y;
    v.z = v.z >= 0.f ? v.z : a * v.z;
    v.w = v.w >= 0.f ? v.w : a * v.w;
    ((float4*)OUT)[idx] = v;
}

// concat=False finalize: OUT[n,c] = prelu(mean_h(AGG[n,h,c]) + bias[c])
__global__ __launch_bounds__(256) void finalize_mean_kernel(
    const float* __restrict__ AGG, const float* __restrict__ bias,
    const float* __restrict__ prelu_a, float* __restrict__ OUT)
{
    const int idx = blockIdx.x * blockDim.x + threadIdx.x;   // float4 units over [N,64]
    if (idx >= NNODES * HIDC / 4) return;
    const int node = idx / (HIDC / 4);
    const int c = (idx % (HIDC / 4)) * 4;
    const float a = *prelu_a;
    const float* base = AGG + (size_t)node * HCAT;
    const float4 v0 = *(const float4*)(base + c);
    const float4 v1 = *(const float4*)(base + HIDC + c);
    float4 v;
    v.x = 0.5f * (v0.x + v1.x) + bias[c + 0];
    v.y = 0.5f * (v0.y + v1.y) + bias[c + 1];
    v.z = 0.5f * (v0.z + v1.z) + bias[c + 2];
    v.w = 0.5f * (v0.w + v1.w) + bias[c + 3];
    v.x = v.x >= 0.f ? v.x : a * v.x;
    v.y = v.y >= 0.f ? v.y : a * v.y;
    v.z = v.z >= 0.f ? v.z : a * v.z;
    v.w = v.w >= 0.f ? v.w : a * v.w;
    *(float4*)(OUT + (size_t)node * HIDC + c) = v;
}

extern "C" void kernel_launch(void* const* d_in, const int* in_sizes, int n_in,
                              void* d_out, int out_size, void* d_ws, size_t ws_size,
                              hipStream_t stream)
{
    const float* x       = (const float*)d_in[0];
    const int*   ei      = (const int*)  d_in[1];
    const float* W1      = (const float*)d_in[2];
    const float* a1s     = (const float*)d_in[3];
    const float* a1d     = (const float*)d_in[4];
    const float* b1      = (const float*)d_in[5];
    const float* W2      = (const float*)d_in[6];
    const float* a2s     = (const float*)d_in[7];
    const float* a2d     = (const float*)d_in[8];
    const float* b2      = (const float*)d_in[9];
    const float* W3      = (const float*)d_in[10];
    const float* a3s     = (const float*)d_in[11];
    const float* a3d     = (const float*)d_in[12];
    const float* b3      = (const float*)d_in[13];
    const float* prelu_a = (const float*)d_in[14];
    const float* Wc      = (const float*)d_in[15];
    const float* bc      = (const float*)d_in[16];
    float* out = (float*)d_out;

    // workspace carve-up (floats)
    float* ws = (float*)d_ws;
    size_t off = 0;
    float* H     = ws + off; off += (size_t)NNODES * HCAT;
    float* AGG   = ws + off; off += (size_t)NNODES * HCAT;
    float* PREV  = ws + off; off += (size_t)NNODES * HCAT;
    float* als   = ws + off; off += (size_t)NNODES * 2;
    float* ald   = ws + off; off += (size_t)NNODES * 2;
    float* maxb  = ws + off; off += (size_t)NNODES * 2;
    float* denom = ws + off; off += (size_t)NNODES * 2;
    float* exbuf = ws + off; off += (size_t)(NEDGES + NNODES) * 2;

    const int ETOT = NEDGES + NNODES;
    const int gN2    = (NNODES * NHEADS + 255) / 256;
    const int gEdge  = (ETOT + 255) / 256;
    const int gEdgeW = (ETOT + 7) / 8;          // wave-per-edge, 8 waves/block
    const int gTile  = ((NNODES / 16) + 7) / 8; // wave-per-16-row-tile, 8 waves/block
    const int gFinC  = (NNODES * HCAT / 4 + 255) / 256;
    const int gFinM  = (NNODES * HIDC / 4 + 255) / 256;

    struct Layer {
        const float* W; const float* as; const float* ad; const float* b; bool concat;
    };
    const Layer layers[3] = {
        { W1, a1s, a1d, b1, true  },
        { W2, a2s, a2d, b2, true  },
        { W3, a3s, a3d, b3, false },
    };

    for (int l = 0; l < 3; ++l) {
        const float* in = (l == 0) ? x : PREV;
        // reset per-layer accumulators (0xFF == "-inf" under the int-max trick)
        hipMemsetAsync(AGG,   0x00, (size_t)NNODES * HCAT * sizeof(float), stream);
        hipMemsetAsync(maxb,  0xFF, (size_t)NNODES * 2 * sizeof(float), stream);
        hipMemsetAsync(denom, 0x00, (size_t)NNODES * 2 * sizeof(float), stream);

        if (l == 0)
            gemm_wmma_f32<INCH, HCAT, 8><<<gTile, 256, 0, stream>>>(in, layers[l].W, nullptr, H, NNODES);
        else
            gemm_wmma_f32<HCAT, HCAT, 8><<<gTile, 256, 0, stream>>>(in, layers[l].W, nullptr, H, NNODES);

        attn_logits_kernel<<<gN2, 256, 0, stream>>>(H, layers[l].as, layers[l].ad, als, ald);
        edge_max_kernel  <<<gEdge, 256, 0, stream>>>(ei, als, ald, maxb);
        edge_exp_kernel  <<<gEdge, 256, 0, stream>>>(ei, als, ald, maxb, exbuf, denom);
        edge_aggr_kernel <<<gEdgeW, 256, 0, stream>>>(ei, H, exbuf, denom, AGG);

        if (layers[l].concat)
            finalize_concat_kernel<<<gFinC, 256, 0, stream>>>(AGG, layers[l].b, prelu_a, PREV);
        else
            finalize_mean_kernel  <<<gFinM, 256, 0, stream>>>(AGG, layers[l].b, prelu_a, PREV);
    }

    // classifier: [N,64] @ [64,12] + bc  (16-wide tile, cols >= 12 masked)
    gemm_wmma_f32<HIDC, NCLS, 1><<<gTile, 256, 0, stream>>>(PREV, Wc, bc, out, NNODES);
    (void)in_sizes; (void)n_in; (void)out_size; (void)ws_size;
}